// CrispComposition_7516192768491
// MI455X (gfx1250) — compile-verified
//
#include <hip/hip_runtime.h>
#include <cstdint>
#include <cstddef>

// CrispComposition: out[b,o] = max_i min(m[b,i], W[i,o])
// (max,min) tropical semiring "GEMM" -> not expressible with WMMA (multiply-add
// ring). CDNA5-specific paths used instead:
//   * GLOBAL_LOAD_ASYNC_TO_LDS_B32/B128 (ASYNCcnt) double-buffered tile staging
//   * s_wait_asynccnt pipelined waits
//   * GLOBAL_ATOMIC_MAX_NUM_F32 (no-return) split-K combine
// Core loop is register-blocked 4x4 VALU min/max (fuses to v_minmax).

#define BATCH 2048
#define IN_F  512
#define OUT_F 256

#define BM 64             // batch rows per block
#define BO 64             // output cols per block
#define KT 32             // k-chunk
#define KSPLIT 2          // split-K factor (grid.z)
#define KRANGE (IN_F / KSPLIT)   // 256 k per block
#define NT (KRANGE / KT)         // 8 k-chunks per block

// ---- CDNA5 async global->LDS copies (tracked by ASYNCcnt) ------------------
__device__ __forceinline__ void async_ld_b32(uint32_t lds_off, uint64_t gaddr) {
  asm volatile("global_load_async_to_lds_b32 %0, %1, off"
               :: "v"(lds_off), "v"(gaddr) : "memory");
}
__device__ __forceinline__ void async_ld_b128(uint32_t lds_off, uint64_t gaddr) {
  asm volatile("global_load_async_to_lds_b128 %0, %1, off"
               :: "v"(lds_off), "v"(gaddr) : "memory");
}
#define WAIT_ASYNC(N) asm volatile("s_wait_asynccnt " #N ::: "memory")

// No-return float max atomic (STOREcnt-tracked, fire-and-forget at L2).
__device__ __forceinline__ void atomic_max_f32(uint64_t gaddr, float v) {
  asm volatile("global_atomic_max_num_f32 %0, %1, off"
               :: "v"(gaddr), "v"(v) : "memory");
}

__global__ __launch_bounds__(256) void crisp_init_kernel(float* __restrict__ out) {
  // 2048*256 floats = 131072 float4; grid 512 x 256 threads
  const size_t i = (size_t)blockIdx.x * blockDim.x + threadIdx.x;
  float4 ninf = make_float4(-__builtin_inff(), -__builtin_inff(),
                            -__builtin_inff(), -__builtin_inff());
  ((float4*)out)[i] = ninf;
}

__global__ __launch_bounds__(256) void crisp_tropical_kernel(
    const float* __restrict__ m,   // [BATCH][IN_F]
    const float* __restrict__ w,   // [IN_F][OUT_F]
    float* __restrict__ out)       // [BATCH][OUT_F]
{
  // Double-buffered tiles. mT is stored K-major (transposed by the async
  // scatter) so the inner loop reads 4 consecutive rows as one b128.
  __shared__ float mT[2][KT][BM];  // 16 KB  mT[buf][k][row] = m[b0+row][k0+k]
  __shared__ float ws[2][KT][BO];  // 16 KB  ws[buf][k][c]   = W[k0+k][o0+c]

  const int tid   = threadIdx.x;
  const int tc    = tid & 15;   // 16 col-groups of 4
  const int tr    = tid >> 4;   // 16 row-groups of 4
  const int o0    = blockIdx.x * BO;
  const int b0    = blockIdx.y * BM;
  const int kbase = blockIdx.z * KRANGE;

  auto issue_tile = [&](int t, int buf) {
    const int k0 = kbase + t * KT;
    // m tile: 64 rows x 32 k = 2048 floats; transpose-scatter, 8 b32 per thread.
    #pragma unroll
    for (int j = 0; j < 8; ++j) {
      const int e   = j * 256 + tid;      // 0..2047
      const int row = e & (BM - 1);
      const int k   = e >> 6;
      const uint64_t g = (uint64_t)(uintptr_t)(m + (size_t)(b0 + row) * IN_F + (k0 + k));
      const uint32_t l = (uint32_t)(uintptr_t)&mT[buf][k][row];
      async_ld_b32(l, g);
    }
    // W tile: 32 k x 64 cols = 2048 floats = 512 16B chunks; 2 b128 per thread.
    #pragma unroll
    for (int j = 0; j < 2; ++j) {
      const int c  = j * 256 + tid;       // 0..511
      const int k  = c >> 4;              // 16 chunks per k-row
      const int c4 = c & 15;
      const uint64_t g = (uint64_t)(uintptr_t)(w + (size_t)(k0 + k) * OUT_F + (o0 + c4 * 4));
      const uint32_t l = (uint32_t)(uintptr_t)&ws[buf][k][c4 * 4];
      async_ld_b128(l, g);
    }
  };

  float acc[4][4];
  #pragma unroll
  for (int r = 0; r < 4; ++r)
    #pragma unroll
    for (int c = 0; c < 4; ++c)
      acc[r][c] = -__builtin_inff();

  issue_tile(0, 0);                 // prologue prefetch (10 async instrs/wave)

  for (int t = 0; t < NT; ++t) {
    const int buf = t & 1;
    if (t + 1 < NT) {
      issue_tile(t + 1, buf ^ 1);   // 10 more in flight
      WAIT_ASYNC(10);               // async loads complete in order => tile t done
    } else {
      WAIT_ASYNC(0);
    }
    __syncthreads();                // tile t visible to every wave

    #pragma unroll
    for (int k = 0; k < KT; ++k) {
      const float4 a  = *(const float4*)&mT[buf][k][tr * 4];
      const float4 wv = *(const float4*)&ws[buf][k][tc * 4];
      const float av[4] = {a.x,  a.y,  a.z,  a.w };
      const float wl[4] = {wv.x, wv.y, wv.z, wv.w};
      #pragma unroll
      for (int r = 0; r < 4; ++r)
        #pragma unroll
        for (int c = 0; c < 4; ++c)
          acc[r][c] = fmaxf(acc[r][c], fminf(av[r], wl[c]));
    }
    __syncthreads();                // compute(t) done before tile t+2 overwrites buf
  }

  // Split-K combine: hardware float max atomics (no return value needed).
  #pragma unroll
  for (int r = 0; r < 4; ++r) {
    const size_t base = (size_t)(b0 + tr * 4 + r) * OUT_F + (o0 + tc * 4);
    #pragma unroll
    for (int c = 0; c < 4; ++c) {
      atomic_max_f32((uint64_t)(uintptr_t)(out + base + c), acc[r][c]);
    }
  }
}

extern "C" void kernel_launch(void* const* d_in, const int* in_sizes, int n_in,
                              void* d_out, int out_size, void* d_ws, size_t ws_size,
                              hipStream_t stream) {
  (void)in_sizes; (void)n_in; (void)d_ws; (void)ws_size; (void)out_size;
  const float* m = (const float*)d_in[0];   // [2048,512] fp32
  const float* w = (const float*)d_in[1];   // [512,256]  fp32
  float* out = (float*)d_out;               // [2048,256] fp32

  // Seed with -inf (deterministic every call; stream order serializes kernels).
  crisp_init_kernel<<<dim3((BATCH * OUT_F / 4) / 256), dim3(256), 0, stream>>>(out);

  dim3 grid(OUT_F / BO, BATCH / BM, KSPLIT);  // (4, 32, 2) = 256 blocks
  dim3 block(256);
  crisp_tropical_kernel<<<grid, block, 0, stream>>>(m, w, out);
}